// TSPM_topKSelection_86440511799909
// MI455X (gfx1250) — compile-verified
//
#include <hip/hip_runtime.h>
#include <hip/hip_bf16.h>
#include <math.h>

// Problem constants (from reference)
#define B_ 32
#define T_ 4096
#define C_ 512
#define H_ 4
#define DH_ 128
#define TOPK_ 10

typedef __attribute__((ext_vector_type(2))) float v2f;
typedef __attribute__((ext_vector_type(8))) float v8f;

// ---------------------------------------------------------------------------
// Kernel A: per batch b, compute
//   q[i]    = qst[b,:] . q_w[i,:] + q_b[i]        (i in [0,512))
//   qh[i]   = q[i] / sqrt(Dh)
//   m[h,c]  = sum_d qh[h*128+d] * k_w[h*128+d, c] (folded score projection)
//   cst[h]  = sum_d qh[h*128+d] * k_b[h*128+d]
// Total work ~17 M MAC across 32 blocks: negligible.
// ---------------------------------------------------------------------------
__global__ __launch_bounds__(256)
void tspm_prep_kernel(const float* __restrict__ qst,
                      const float* __restrict__ q_w,
                      const float* __restrict__ q_b,
                      const float* __restrict__ k_w,
                      const float* __restrict__ k_b,
                      float* __restrict__ m_out,      // [B][H][C]
                      float* __restrict__ cst_out)    // [B][H]
{
    const int b   = blockIdx.x;
    const int tid = threadIdx.x;

    __shared__ float qst_s[C_];
    __shared__ float qh_s[C_];

    for (int i = tid; i < C_; i += 256) qst_s[i] = qst[b * C_ + i];
    __syncthreads();

    const float scale = 0.08838834764831845f; // 1/sqrt(128)
    for (int i = tid; i < C_; i += 256) {
        float acc = q_b[i];
        const float* wrow = q_w + (size_t)i * C_;
        #pragma unroll 4
        for (int j = 0; j < C_; ++j) acc = fmaf(qst_s[j], wrow[j], acc);
        qh_s[i] = acc * scale;
    }
    __syncthreads();

    // m[h][c]: consecutive tid -> consecutive c -> coalesced k_w reads
    for (int o = tid; o < H_ * C_; o += 256) {
        const int h = o >> 9;
        const int c = o & (C_ - 1);
        const float* q = qh_s + h * DH_;
        float acc = 0.0f;
        #pragma unroll 4
        for (int d = 0; d < DH_; ++d)
            acc = fmaf(q[d], k_w[(size_t)(h * DH_ + d) * C_ + c], acc);
        m_out[((size_t)b * H_ + h) * C_ + c] = acc;
    }
    if (tid < H_) {
        float acc = 0.0f;
        for (int d = 0; d < DH_; ++d)
            acc = fmaf(qh_s[tid * DH_ + d], k_b[tid * DH_ + d], acc);
        cst_out[b * H_ + tid] = acc;
    }
}

// ---------------------------------------------------------------------------
// Kernel B: scores[b,h,t] = m[b,h,:] . visual[b,t,:] + cst[b,h]
// via V_WMMA_F32_16X16X4_F32.  One wave per 16-row t-tile, 8 waves/block.
// A (16x4 f32): lane l<16 holds row (t0+l), K = k,k+1 ; lane l>=16 same row
//               group with K = k+2,k+3  -> float2 global load per K-step.
// B (4x16 f32): column n = head (rows 4..15 of staged m are zero so the
//               fragment load is uniform and EXEC stays all-ones).
// C/D (16x16 f32): VGPR v, lane l -> t = t0 + v + 8*(l>=16), head = l&15.
// LDS m stage padded to pitch 514 floats: bank = (2n + k) % 64, conflict-free
// b64 loads across the half-wave.
// ---------------------------------------------------------------------------
#define MPITCH 514

__global__ __launch_bounds__(256)
void tspm_scores_wmma_kernel(const float* __restrict__ visual,
                             const float* __restrict__ m_in,   // [B][H][C]
                             const float* __restrict__ cst_in, // [B][H]
                             float* __restrict__ scores)       // [B][H][T]
{
    const int b    = blockIdx.y;
    const int tid  = threadIdx.x;
    const int wave = tid >> 5;
    const int lane = tid & 31;
    const int lrow = lane & 15;       // A row / D column index
    const int half = lane >> 4;       // 0: K+{0,1} / M 0-7,  1: K+{2,3} / M 8-15

    __shared__ float m_lds[16 * MPITCH];

    // Zero the 16x(padded C) stage, then fill rows 0..3 with m[b,h,:].
    for (int i = tid; i < 16 * MPITCH; i += 256) m_lds[i] = 0.0f;
    __syncthreads();
    for (int i = tid; i < H_ * C_; i += 256) {
        const int h = i >> 9;
        const int c = i & (C_ - 1);
        m_lds[h * MPITCH + c] = m_in[((size_t)b * H_ + h) * C_ + c];
    }
    __syncthreads();

    const int t0 = blockIdx.x * 128 + wave * 16;

    const float* aptr = visual + ((size_t)b * T_ + t0 + lrow) * C_ + 2 * half;
    const float* bptr = &m_lds[lrow * MPITCH + 2 * half];

    v8f acc = {};
    #pragma unroll 8
    for (int k = 0; k < C_; k += 4) {
        v2f a  = *(const v2f*)(aptr + k);   // global_load_b64
        v2f bb = *(const v2f*)(bptr + k);   // ds_load_b64 (conflict-free)
        acc = __builtin_amdgcn_wmma_f32_16x16x4_f32(
                  false, a, false, bb, (short)0, acc, false, false);
    }

    // Epilogue (divergence OK here, WMMAs are done): heads live in cols 0..3.
    if (lrow < H_) {
        const float cst = cst_in[b * H_ + lrow];
        float* srow = scores + ((size_t)b * H_ + lrow) * T_;
        #pragma unroll
        for (int v = 0; v < 8; ++v)
            srow[t0 + v + 8 * half] = acc[v] + cst;
    }
}

// ---------------------------------------------------------------------------
// Kernel C: per batch b: softmax over T per head, mean over heads,
// top-10 by iterative block argmax, sort indices ascending, gather rows.
// ---------------------------------------------------------------------------
__global__ __launch_bounds__(256)
void tspm_topk_gather_kernel(const float* __restrict__ scores, // [B][H][T]
                             const float* __restrict__ audio,
                             const float* __restrict__ pa,
                             const float* __restrict__ pv,
                             float* __restrict__ out)           // 3 x [B][10][C]
{
    const int b   = blockIdx.x;
    const int tid = threadIdx.x;

    __shared__ float tw[T_];          // 16 KB mean-softmax weights
    __shared__ float redf[256];
    __shared__ int   redi[256];
    __shared__ int   sel[TOPK_];

    for (int i = tid; i < T_; i += 256) tw[i] = 0.0f;
    __syncthreads();

    for (int h = 0; h < H_; ++h) {
        const float* s = scores + ((size_t)b * H_ + h) * T_;
        float vals[16];
        float lm = -INFINITY;
        #pragma unroll
        for (int j = 0; j < 16; ++j) {
            vals[j] = s[tid + j * 256];
            lm = fmaxf(lm, vals[j]);
        }
        redf[tid] = lm; __syncthreads();
        for (int st = 128; st > 0; st >>= 1) {
            if (tid < st) redf[tid] = fmaxf(redf[tid], redf[tid + st]);
            __syncthreads();
        }
        const float mx = redf[0]; __syncthreads();

        float ls = 0.0f;
        #pragma unroll
        for (int j = 0; j < 16; ++j) { vals[j] = __expf(vals[j] - mx); ls += vals[j]; }
        redf[tid] = ls; __syncthreads();
        for (int st = 128; st > 0; st >>= 1) {
            if (tid < st) redf[tid] += redf[tid + st];
            __syncthreads();
        }
        const float inv = 0.25f / redf[0]; __syncthreads();

        #pragma unroll
        for (int j = 0; j < 16; ++j) tw[tid + j * 256] += vals[j] * inv;
        __syncthreads();
    }

    // Iterative top-10 (block argmax; ties -> lowest index)
    for (int it = 0; it < TOPK_; ++it) {
        float lm = -INFINITY; int li = T_;
        #pragma unroll
        for (int j = 0; j < 16; ++j) {
            const int   i2 = tid + j * 256;
            const float v  = tw[i2];
            if (v > lm || (v == lm && i2 < li)) { lm = v; li = i2; }
        }
        redf[tid] = lm; redi[tid] = li; __syncthreads();
        for (int st = 128; st > 0; st >>= 1) {
            if (tid < st) {
                if (redf[tid + st] > redf[tid] ||
                    (redf[tid + st] == redf[tid] && redi[tid + st] < redi[tid])) {
                    redf[tid] = redf[tid + st];
                    redi[tid] = redi[tid + st];
                }
            }
            __syncthreads();
        }
        if (tid == 0) { sel[it] = redi[0]; tw[redi[0]] = -INFINITY; }
        __syncthreads();
    }

    if (tid == 0) { // sort 10 indices ascending
        for (int i = 1; i < TOPK_; ++i) {
            int v = sel[i], j = i - 1;
            while (j >= 0 && sel[j] > v) { sel[j + 1] = sel[j]; --j; }
            sel[j + 1] = v;
        }
    }
    __syncthreads();

    const size_t sec = (size_t)B_ * TOPK_ * C_; // 163840 per output tensor
    for (int o = tid; o < TOPK_ * C_; o += 256) {
        const int j = o >> 9;
        const int c = o & (C_ - 1);
        const size_t src = ((size_t)b * T_ + sel[j]) * C_ + c;
        const size_t dst = ((size_t)b * TOPK_ + j) * C_ + c;
        out[dst]           = audio[src];
        out[sec + dst]     = pa[src];
        out[2 * sec + dst] = pv[src];
    }
}

// ---------------------------------------------------------------------------
// Launch: prep -> WMMA scores (streams the 256 MB of visual once) -> top-k.
// Workspace layout (floats): m[B*H*C]=65536 | cst[B*H]=128 | scores[B*H*T].
// ---------------------------------------------------------------------------
extern "C" void kernel_launch(void* const* d_in, const int* in_sizes, int n_in,
                              void* d_out, int out_size, void* d_ws, size_t ws_size,
                              hipStream_t stream) {
    const float* audio  = (const float*)d_in[0];
    const float* visual = (const float*)d_in[1];
    const float* pa     = (const float*)d_in[2];
    const float* pv     = (const float*)d_in[3];
    const float* qst    = (const float*)d_in[4];
    const float* q_w    = (const float*)d_in[5];
    const float* q_b    = (const float*)d_in[6];
    const float* k_w    = (const float*)d_in[7];
    const float* k_b    = (const float*)d_in[8];
    float* out = (float*)d_out;

    float* ws_f   = (float*)d_ws;
    float* m_ws   = ws_f;                       // 65536 floats
    float* cst_ws = ws_f + (size_t)B_ * H_ * C_;        // 128 floats
    float* sc_ws  = cst_ws + 128;               // B*H*T floats

    tspm_prep_kernel<<<dim3(B_), dim3(256), 0, stream>>>(
        qst, q_w, q_b, k_w, k_b, m_ws, cst_ws);

    tspm_scores_wmma_kernel<<<dim3(T_ / 128, B_), dim3(256), 0, stream>>>(
        visual, m_ws, cst_ws, sc_ws);

    tspm_topk_gather_kernel<<<dim3(B_), dim3(256), 0, stream>>>(
        sc_ws, audio, pa, pv, out);
}